// WeightedAttentionPooling_53274774340079
// MI455X (gfx1250) — compile-verified
//
#include <hip/hip_runtime.h>
#include <hip/hip_bf16.h>

typedef __attribute__((ext_vector_type(2))) float        v2f;
typedef __attribute__((ext_vector_type(8))) float        v8f;
typedef __attribute__((ext_vector_type(4))) unsigned int v4u;
typedef __attribute__((ext_vector_type(8))) int          v8i;
typedef __attribute__((ext_vector_type(4))) int          v4i;

#define DD 128            // feature dim (known from reference)
#define LDS_STRIDE 132    // 128 floats/row + 4-float pad (TDM pad feature)

// ---------------------------------------------------------------------------
// Kernel 1: per-node gate projection.
//   er[n] = exp(x[n]·Wg[0:128] + bg)
//   ec[n] = pos[n] * exp(x[n]·Wg[128:256])
// One wave (32 lanes) per node; each lane handles 4 contiguous floats.
// ---------------------------------------------------------------------------
__global__ void node_gate_kernel(const float* __restrict__ x,
                                 const float* __restrict__ pos,
                                 const float* __restrict__ Wg,
                                 const float* __restrict__ bg,
                                 float* __restrict__ er,
                                 float* __restrict__ ec,
                                 int N_) {
    int wave = threadIdx.x >> 5;
    int lane = threadIdx.x & 31;
    int n = blockIdx.x * 8 + wave;
    if (n >= N_) return;

    const float4* xr = (const float4*)(x + (size_t)n * DD);
    const float4* wt = (const float4*)(Wg);          // Wg[0:128]
    const float4* wb = (const float4*)(Wg + DD);     // Wg[128:256]

    float4 xv  = xr[lane];
    float4 wtv = wt[lane];
    float4 wbv = wb[lane];

    float gt = xv.x * wtv.x + xv.y * wtv.y + xv.z * wtv.z + xv.w * wtv.w;
    float gb = xv.x * wbv.x + xv.y * wbv.y + xv.z * wbv.z + xv.w * wbv.w;

    #pragma unroll
    for (int off = 16; off >= 1; off >>= 1) {
        gt += __shfl_xor(gt, off, 32);
        gb += __shfl_xor(gb, off, 32);
    }
    if (lane == 0) {
        er[n] = __expf(gt + bg[0]);
        ec[n] = pos[n] * __expf(gb);
    }
}

// ---------------------------------------------------------------------------
// Kernel 2: per-edge unnormalized weight + rowsum accumulation.
//   a_e = er[row]*ec[col];  rowsum[row] += a_e
// ---------------------------------------------------------------------------
__global__ void edge_gate_kernel(const long long* __restrict__ ei,
                                 const float* __restrict__ er,
                                 const float* __restrict__ ec,
                                 float* __restrict__ a_ws,
                                 float* __restrict__ rowsum,
                                 int E_) {
    int e = blockIdx.x * blockDim.x + threadIdx.x;
    if (e >= E_) return;
    int row = (int)ei[e];
    int col = (int)ei[(size_t)E_ + e];
    float a = er[row] * ec[col];
    a_ws[e] = a;
    atomicAdd(&rowsum[row], a);
}

// ---------------------------------------------------------------------------
// Kernel 3: inv = 1/rowsum + 1e-16 (in place)
// ---------------------------------------------------------------------------
__global__ void inv_kernel(float* __restrict__ rowsum, int N_) {
    int n = blockIdx.x * blockDim.x + threadIdx.x;
    if (n < N_) rowsum[n] = 1.0f / rowsum[n] + 1e-16f;
}

// ---------------------------------------------------------------------------
// Kernel 4: per-edge scatter. One wave per edge:
//   alpha = a_ws[e] * inv[row]
//   S[row]        += alpha            (lane 0)
//   Y[row][0:128] += alpha * x[col]   (lane d: 4 floats each, float4 load)
// x and Y are both ~5 MB => L2-resident atomic traffic.
// ---------------------------------------------------------------------------
__global__ void edge_scatter_kernel(const long long* __restrict__ ei,
                                    const float* __restrict__ x,
                                    const float* __restrict__ a_ws,
                                    const float* __restrict__ inv,
                                    float* __restrict__ S,
                                    float* __restrict__ Y,
                                    int E_) {
    int wid  = (int)(((size_t)blockIdx.x * blockDim.x + threadIdx.x) >> 5);
    int lane = threadIdx.x & 31;
    if (wid >= E_) return;
    int row = (int)ei[wid];
    int col = (int)ei[(size_t)E_ + wid];
    float alpha = a_ws[wid] * inv[row];
    if (lane == 0) atomicAdd(&S[row], alpha);

    const float4* xc = (const float4*)(x + (size_t)col * DD);
    float4 v = xc[lane];
    float* yr = Y + (size_t)row * DD + lane * 4;
    atomicAdd(yr + 0, alpha * v.x);
    atomicAdd(yr + 1, alpha * v.y);
    atomicAdd(yr + 2, alpha * v.z);
    atomicAdd(yr + 3, alpha * v.w);
}

// ---------------------------------------------------------------------------
// TDM descriptor helper: 2D tile load, 16 rows x 128 f32 elements, with
// 4-DWORD padding after every 128 DWORDs so LDS rows land on a 132-float
// stride (bank-conflict-free for the stride-per-row A-fragment reads).
//   group0: count=1 | lds_addr | global_addr | type=2
//   group1: data_size=4B, pad_enable, pad_interval=128dw(code 6),
//           pad_amount=4dw(code 3); tensor_dim0=128, tensor_dim1=16,
//           tile_dim0=128, tile_dim1=16, tensor_dim0_stride=128
// Toolchain here exposes the 6-arg builtin:
//   (uint32x4 g0, int32x8 g1, int32x4 g2, int32x4 g3, int32x8 pad, i32 cpol)
// ---------------------------------------------------------------------------
__device__ __forceinline__ void tdm_load_tile16x128(unsigned int lds_byte_off,
                                                    const float* gsrc) {
    unsigned long long ga = (unsigned long long)(const void*)gsrc;
    v4u g0 = { 1u,                              // count=1, is_restore=0
               lds_byte_off,                    // lds_addr
               (unsigned int)ga,                // global_addr[31:0]
               (unsigned int)((ga >> 32) & 0x01FFFFFFu) | (2u << 30) }; // +type=2
    v8i g1 = { (int)((2u << 16) | (1u << 20) | (6u << 22) | (3u << 25)),
               (int)(128u << 16),               // tensor_dim0 = 128 (low16 -> bits 63:48)
               (int)(16u << 16),                // tensor_dim1 = 16  (low16 -> bits 95:80)
               (int)(128u << 16),               // tile_dim0   = 128 (bits 127:112)
               16,                              // tile_dim1   = 16  (bits 143:128)
               128,                             // tensor_dim0_stride = 128 (bits 191:160)
               0, 0 };
    v4i z4 = { 0, 0, 0, 0 };
    v8i z8 = { 0, 0, 0, 0, 0, 0, 0, 0 };
    __builtin_amdgcn_tensor_load_to_lds(g0, g1, z4, z4, z8, 0);
}

// ---------------------------------------------------------------------------
// Kernel 5: final fp32 WMMA GEMM.
//   out[m][n] = S[m]*(x[m]@Wm_top)[n] + (Y[m]@Wm_bot)[n] + S[m]*bm[n]
// Tile: 16x16 per wave, two K=128 loops of V_WMMA_F32_16X16X4_F32.
// Block = 256 threads = 8 waves = the 8 N-tiles of one 16-row M-tile;
// grid = 625 blocks (10000/16). The shared 16x128 x-tile and Y-tile are
// DMA'd into LDS once per block by the Tensor Data Mover (wave 0 issues,
// waits on TENSORcnt, then the workgroup barrier releases everyone).
// A frag (ISA 7.12.2, 32-bit 16x4): lanes 0-15 -> K=k0,k0+1; lanes 16-31 ->
// K=k0+2,k0+3; M = lane&15.  C/D: VGPR j -> M = j + 8*(lane>=16), N = lane&15.
// ---------------------------------------------------------------------------
__global__ void out_gemm_kernel(const float* __restrict__ x,
                                const float* __restrict__ Wm,
                                const float* __restrict__ bm,
                                const float* __restrict__ S,
                                const float* __restrict__ Y,
                                float* __restrict__ out) {
    __shared__ float smem[2 * 16 * LDS_STRIDE];   // x-tile | Y-tile (padded rows)

    int wave   = threadIdx.x >> 5;      // tile_n = 0..7
    int lane   = threadIdx.x & 31;
    int tile_m = blockIdx.x;            // 0..624
    int half   = lane >> 4;             // 0 or 1
    int mloc   = lane & 15;
    int n      = wave * 16 + mloc;      // B/C/D column this lane holds

    if (wave == 0) {
        unsigned int lds0 = (unsigned int)(unsigned long long)(const void*)&smem[0];
        unsigned int lds1 = (unsigned int)(unsigned long long)(const void*)&smem[16 * LDS_STRIDE];
        tdm_load_tile16x128(lds0, x + (size_t)tile_m * 16 * DD);
        tdm_load_tile16x128(lds1, Y + (size_t)tile_m * 16 * DD);
        __builtin_amdgcn_s_wait_tensorcnt(0);
    }
    __syncthreads();

    const float* xs = &smem[0];
    const float* ys = &smem[16 * LDS_STRIDE];

    v8f c1 = {};   // x @ Wm_top   (row-scale by S applied in epilogue)
    v8f c2 = {};   // Y @ Wm_bot
    #pragma unroll 4
    for (int kk = 0; kk < DD; kk += 4) {
        int kb = kk + half * 2;         // this lane's first K
        v2f a, b;

        float2 ax = *(const float2*)&xs[mloc * LDS_STRIDE + kb];
        a[0] = ax.x;  a[1] = ax.y;
        b[0] = Wm[(size_t)kb * DD + n];
        b[1] = Wm[(size_t)(kb + 1) * DD + n];
        c1 = __builtin_amdgcn_wmma_f32_16x16x4_f32(false, a, false, b,
                                                   (short)0, c1, false, false);

        float2 ay = *(const float2*)&ys[mloc * LDS_STRIDE + kb];
        a[0] = ay.x;  a[1] = ay.y;
        b[0] = Wm[(size_t)(DD + kb) * DD + n];
        b[1] = Wm[(size_t)(DD + kb + 1) * DD + n];
        c2 = __builtin_amdgcn_wmma_f32_16x16x4_f32(false, a, false, b,
                                                   (short)0, c2, false, false);
    }

    float bmn = bm[n];
    #pragma unroll
    for (int j = 0; j < 8; ++j) {
        int mrow = tile_m * 16 + j + half * 8;
        out[(size_t)mrow * DD + n] = S[mrow] * (c1[j] + bmn) + c2[j];
    }
}

// ---------------------------------------------------------------------------
// Launch. Inputs: x[N*128], pos[N], Wg[256], bg[1], Wm[256*128], bm[128],
// edge_index[2*E] (int64). Output: out[N*128] f32.
// Workspace layout (floats): [rowsum N][S N][Y N*128][er N][ec N][a_ws E]
// = (132N + E)*4 bytes ~ 7.9 MB.
// ---------------------------------------------------------------------------
extern "C" void kernel_launch(void* const* d_in, const int* in_sizes, int n_in,
                              void* d_out, int out_size, void* d_ws, size_t ws_size,
                              hipStream_t stream) {
    const float*     x   = (const float*)d_in[0];
    const float*     pos = (const float*)d_in[1];
    const float*     Wg  = (const float*)d_in[2];
    const float*     bg  = (const float*)d_in[3];
    const float*     Wm  = (const float*)d_in[4];
    const float*     bm  = (const float*)d_in[5];
    const long long* ei  = (const long long*)d_in[6];

    const int N_ = in_sizes[1];          // 10000
    const int E_ = in_sizes[6] / 2;      // 640000

    float* p      = (float*)d_ws;
    float* rowsum = p;                   // N  (becomes inv in place)
    float* S      = p + N_;              // N
    float* Y      = p + 2 * N_;          // N*128
    float* er     = p + 2 * N_ + (size_t)N_ * DD;
    float* ec     = er + N_;
    float* a_ws   = ec + N_;

    float* out = (float*)d_out;

    // zero the accumulators (rowsum, S, Y are contiguous)
    (void)hipMemsetAsync(d_ws, 0, (size_t)(2 * N_ + (size_t)N_ * DD) * sizeof(float), stream);

    // 1) node gate projections: 8 waves/block, one node per wave
    node_gate_kernel<<<(N_ + 7) / 8, 256, 0, stream>>>(x, pos, Wg, bg, er, ec, N_);

    // 2) edge gate weights + rowsum
    edge_gate_kernel<<<(E_ + 255) / 256, 256, 0, stream>>>(ei, er, ec, a_ws, rowsum, E_);

    // 3) inverse
    inv_kernel<<<(N_ + 255) / 256, 0ULL ? 0 : 256, 0, stream>>>(rowsum, N_);

    // 4) weighted scatter: one wave per edge
    {
        long long waves = E_;
        long long blocks = (waves * 32 + 255) / 256;
        edge_scatter_kernel<<<(int)blocks, 256, 0, stream>>>(ei, x, a_ws, rowsum, S, Y, E_);
    }

    // 5) fp32 WMMA GEMM with TDM-staged A tiles: out = [S*x | Y] @ Wm + S*bm
    out_gemm_kernel<<<N_ / 16, 256, 0, stream>>>(x, Wm, bm, S, Y, out);
}